// WECT_61924838474102
// MI455X (gfx1250) — compile-verified
//
#include <hip/hip_runtime.h>
#include <cstdint>
#include <cstddef>

// ---------------------------------------------------------------------------
// WECT on MI455X (gfx1250).
//   dirs (32,3) f32 | v_coords (K0,3) f32 | v_weights (K0,) f32
//   edge_verts (K1,2) i32 | edge_weights (K1,) f32
//   tri_verts (K2,3) i32  | tri_weights (K2,) f32 | num_heights (1,) i32
// out: (32, H) f32 cumsum of signed histogram.
//
// Strategy: WMMA f32 16x16x4 for the height GEMM; uint8 bin table [K0][32]
// in workspace (fits in the 192MB L2); fused LDS-atomic histogram; cumsum.
// ---------------------------------------------------------------------------

#define NDIR   32
#define MAXH   256

typedef float v2f __attribute__((ext_vector_type(2)));
typedef float v8f __attribute__((ext_vector_type(8)));

// ------------------------- init: zero accumulators -------------------------
__global__ void wect_init(float* gdiff, unsigned int* gmax2) {
    int i = blockIdx.x * blockDim.x + threadIdx.x;
    if (i < NDIR * MAXH) gdiff[i] = 0.0f;
    if (i == 0) *gmax2 = 0u;
}

// ------------------------- normalize directions ----------------------------
__global__ void wect_norm_dirs(const float* __restrict__ dirs,
                               float* __restrict__ wdirs, int D) {
    int d = blockIdx.x * blockDim.x + threadIdx.x;
    if (d >= D) return;
    float x = dirs[3 * d + 0], y = dirs[3 * d + 1], z = dirs[3 * d + 2];
    float n = fmaxf(sqrtf(x * x + y * y + z * z), 1e-12f);
    wdirs[3 * d + 0] = x / n;
    wdirs[3 * d + 1] = y / n;
    wdirs[3 * d + 2] = z / n;
}

// ------------------------- max ||v||^2 reduction ---------------------------
__global__ void wect_max_norm(const float* __restrict__ vc, int K0,
                              unsigned int* gmax2) {
    int stride = gridDim.x * blockDim.x;
    float m = 0.0f;
    for (int v = blockIdx.x * blockDim.x + threadIdx.x; v < K0; v += stride) {
        float x = vc[3 * v + 0], y = vc[3 * v + 1], z = vc[3 * v + 2];
        m = fmaxf(m, x * x + y * y + z * z);
    }
    // wave32 reduction
    #pragma unroll
    for (int off = 16; off > 0; off >>= 1)
        m = fmaxf(m, __shfl_xor(m, off));
    if ((threadIdx.x & 31) == 0)
        atomicMax((int*)gmax2, __float_as_int(m));   // valid: m >= 0
}

// --------- one-thread finalize: hoist sqrt + divide out of hot loop --------
__global__ void wect_scale(const unsigned int* __restrict__ gmax2,
                           const int* __restrict__ hptr,
                           float* __restrict__ sc) {
    float maxh = fmaxf(sqrtf(__int_as_float((int)*gmax2)), 1e-12f);
    int H = *hptr;
    sc[0] = maxh;
    sc[1] = (float)(H - 1) / (2.0f * maxh);   // scale
    sc[2] = (float)(H - 1);                   // clamp hi
}

// ---------------- WMMA height GEMM + quantize to uint8 bins ----------------
// One wave handles tiles of 16 vertices x 32 dirs via two
// v_wmma_f32_16x16x4_f32 (K padded 3->4); grid-stride tile loop with B and
// scale hoisted. Bins transposed through LDS, stored coalesced [K0][32] u8.
__global__ void wect_index(const float* __restrict__ vc,
                           const float* __restrict__ wdirs,
                           const float* __restrict__ sc,
                           unsigned char* __restrict__ vidx, int K0) {
    __shared__ unsigned char sm[8 * 512];          // 512 B per wave (8 waves)
    const float maxh  = sc[0];
    const float scale = sc[1];
    const float hm1   = sc[2];

    const int lane = threadIdx.x & 31;
    const int wave = threadIdx.x >> 5;
    const int nloc = lane & 15;
    const int hi   = lane >> 4;                    // lane half selects K pair

    // B (4x16 f32): VGPR0 = K0 (lanes 0-15) / K2 (lanes 16-31); VGPR1 = K1/K3
    const float* dp0 = wdirs + 3 * nloc;
    const float* dp1 = wdirs + 3 * (nloc + 16);
    float d0x = dp0[0], d0y = dp0[1], d0z = dp0[2];
    float d1x = dp1[0], d1y = dp1[1], d1z = dp1[2];
    v2f b0, b1;
    b0.x = hi ? d0z : d0x;
    b0.y = hi ? 0.0f : d0y;
    b1.x = hi ? d1z : d1x;
    b1.y = hi ? 0.0f : d1y;

    unsigned char* ws = sm + wave * 512;
    const int nTiles = (K0 + 15) >> 4;
    const int waveId = blockIdx.x * 8 + wave;
    const int nWaves = gridDim.x * 8;

    for (int tile = waveId; tile < nTiles; tile += nWaves) {
        const int base = tile << 4;

        // A (16x4 f32): lanes 0-15 hold (x,y), lanes 16-31 hold (z,0).
        // Load all 3 coords unconditionally -> cndmask select, no EXEC branch.
        int row = base + nloc;
        if (row >= K0) row = K0 - 1;               // clamp tail (discarded)
        const float* vp = vc + (size_t)row * 3;
        float vx = vp[0], vy = vp[1], vz = vp[2];
        v2f a;
        a.x = hi ? vz : vx;
        a.y = hi ? 0.0f : vy;

        v8f c0 = {};
        v8f c1 = {};
        c0 = __builtin_amdgcn_wmma_f32_16x16x4_f32(false, a, false, b0,
                                                   (short)0, c0, false, false);
        c1 = __builtin_amdgcn_wmma_f32_16x16x4_f32(false, a, false, b1,
                                                   (short)0, c1, false, false);

        // C layout: VGPR r -> M = r + 8*hi, N = nloc (c0: dirs 0-15, c1: 16-31)
        #pragma unroll
        for (int r = 0; r < 8; ++r) {
            int rr = r + (hi << 3);
            float f0 = ceilf((maxh + c0[r]) * scale);
            f0 = fminf(fmaxf(f0, 0.0f), hm1);
            ws[rr * 32 + nloc] = (unsigned char)(int)f0;
            float f1 = ceilf((maxh + c1[r]) * scale);
            f1 = fminf(fmaxf(f1, 0.0f), hm1);
            ws[rr * 32 + 16 + nloc] = (unsigned char)(int)f1;
        }
        // Same-wave DS ops are in-order; wait + compiler barrier before reread.
        asm volatile("s_wait_dscnt 0" ::: "memory");

        // Coalesced write-back: each lane stores 16 B (half a vertex row).
        if (base + (lane >> 1) < K0) {
            uint4 val = *(const uint4*)(ws + lane * 16);
            *(uint4*)(vidx + (size_t)base * 32 + lane * 16) = val;
        }
    }
}

// ------------------------- histogram helpers -------------------------------
__device__ __forceinline__ uint32_t maxu8x4(uint32_t a, uint32_t b) {
    uint32_t r = 0;
    #pragma unroll
    for (int i = 0; i < 4; ++i) {
        uint32_t x = (a >> (8 * i)) & 0xffu;
        uint32_t y = (b >> (8 * i)) & 0xffu;
        r |= (x > y ? x : y) << (8 * i);
    }
    return r;
}

__device__ __forceinline__ void accum_bins(float* h, int H,
                                           uint4 lo, uint4 hi, float w) {
    uint32_t wd[8] = {lo.x, lo.y, lo.z, lo.w, hi.x, hi.y, hi.z, hi.w};
    #pragma unroll
    for (int i = 0; i < 8; ++i) {
        #pragma unroll
        for (int b = 0; b < 4; ++b) {
            int d   = i * 4 + b;
            int bin = (int)((wd[i] >> (8 * b)) & 0xffu);
            atomicAdd(&h[d * H + bin], w);         // ds_add_f32
        }
    }
}

// ------------- fused vertex/edge/triangle LDS histogram --------------------
__global__ void wect_hist(const unsigned char* __restrict__ vidx,
                          const float* __restrict__ vw,
                          const int* __restrict__ ev,
                          const float* __restrict__ ew,
                          const int* __restrict__ tv,
                          const float* __restrict__ tw,
                          const int* __restrict__ hptr,
                          float* __restrict__ gdiff,
                          int K0, int K1, int K2) {
    __shared__ float h[NDIR * MAXH];               // 32 KB
    const int H   = *hptr;
    const int tid = threadIdx.x;
    for (int i = tid; i < NDIR * MAXH; i += blockDim.x) h[i] = 0.0f;
    __syncthreads();

    const int stride = gridDim.x * blockDim.x;
    const int g = blockIdx.x * blockDim.x + tid;

    // vertices: +w at own bins
    for (int v = g; v < K0; v += stride) {
        const uint4* p = (const uint4*)(vidx + (size_t)v * 32);
        accum_bins(h, H, p[0], p[1], vw[v]);
    }
    // edges: -w at per-dir max of 2 vertex bins
    for (int e = g; e < K1; e += stride) {
        int va = ev[2 * e], vb = ev[2 * e + 1];
        const uint4* pa = (const uint4*)(vidx + (size_t)va * 32);
        const uint4* pb = (const uint4*)(vidx + (size_t)vb * 32);
        uint4 a0 = pa[0], a1 = pa[1], b0 = pb[0], b1 = pb[1];
        uint4 m0 = make_uint4(maxu8x4(a0.x, b0.x), maxu8x4(a0.y, b0.y),
                              maxu8x4(a0.z, b0.z), maxu8x4(a0.w, b0.w));
        uint4 m1 = make_uint4(maxu8x4(a1.x, b1.x), maxu8x4(a1.y, b1.y),
                              maxu8x4(a1.z, b1.z), maxu8x4(a1.w, b1.w));
        accum_bins(h, H, m0, m1, -ew[e]);
    }
    // triangles: +w at per-dir max of 3 vertex bins
    for (int t = g; t < K2; t += stride) {
        int va = tv[3 * t], vb = tv[3 * t + 1], vcx = tv[3 * t + 2];
        const uint4* pa = (const uint4*)(vidx + (size_t)va * 32);
        const uint4* pb = (const uint4*)(vidx + (size_t)vb * 32);
        const uint4* pc = (const uint4*)(vidx + (size_t)vcx * 32);
        uint4 a0 = pa[0], a1 = pa[1], b0 = pb[0], b1 = pb[1];
        uint4 c0 = pc[0], c1 = pc[1];
        uint4 m0 = make_uint4(maxu8x4(maxu8x4(a0.x, b0.x), c0.x),
                              maxu8x4(maxu8x4(a0.y, b0.y), c0.y),
                              maxu8x4(maxu8x4(a0.z, b0.z), c0.z),
                              maxu8x4(maxu8x4(a0.w, b0.w), c0.w));
        uint4 m1 = make_uint4(maxu8x4(maxu8x4(a1.x, b1.x), c1.x),
                              maxu8x4(maxu8x4(a1.y, b1.y), c1.y),
                              maxu8x4(maxu8x4(a1.z, b1.z), c1.z),
                              maxu8x4(maxu8x4(a1.w, b1.w), c1.w));
        accum_bins(h, H, m0, m1, tw[t]);
    }

    __syncthreads();
    for (int i = tid; i < NDIR * H; i += blockDim.x) {
        float v = h[i];
        if (v != 0.0f) atomicAdd(&gdiff[i], v);
    }
}

// ------------------------- cumsum over heights -----------------------------
__global__ void wect_scan(const float* __restrict__ gdiff,
                          const int* __restrict__ hptr,
                          float* __restrict__ out) {
    const int H = *hptr;
    int d = threadIdx.x;
    if (d >= NDIR) return;
    float acc = 0.0f;
    for (int j = 0; j < H; ++j) {
        acc += gdiff[d * H + j];
        out[d * H + j] = acc;
    }
}

// ---------------------------------------------------------------------------
extern "C" void kernel_launch(void* const* d_in, const int* in_sizes, int n_in,
                              void* d_out, int out_size, void* d_ws,
                              size_t ws_size, hipStream_t stream) {
    const float* dirs = (const float*)d_in[0];
    const float* vc   = (const float*)d_in[1];
    const float* vw   = (const float*)d_in[2];
    const int*   ev   = (const int*)d_in[3];
    const float* ew   = (const float*)d_in[4];
    const int*   tv   = (const int*)d_in[5];
    const float* tw   = (const float*)d_in[6];
    const int*   hptr = (const int*)d_in[7];

    const int K0 = in_sizes[2];        // vertices
    const int K1 = in_sizes[4];        // edges
    const int K2 = in_sizes[6];        // triangles
    const int D  = in_sizes[0] / 3;    // 32 directions

    // workspace layout
    char* ws = (char*)d_ws;
    float*         gdiff = (float*)(ws + 0);                 // 32*256 f32
    unsigned int*  gmax2 = (unsigned int*)(ws + 32768);      // max ||v||^2
    float*         scbuf = (float*)(ws + 32896);             // maxh/scale/hm1
    float*         wdirs = (float*)(ws + 33024);             // normalized dirs
    unsigned char* vidx  = (unsigned char*)(ws + 33536);     // [K0][32] u8

    wect_init<<<(NDIR * MAXH + 256) / 256, 256, 0, stream>>>(gdiff, gmax2);
    wect_norm_dirs<<<(D + 31) / 32, 32, 0, stream>>>(dirs, wdirs, D);
    wect_max_norm<<<256, 256, 0, stream>>>(vc, K0, gmax2);
    wect_scale<<<1, 1, 0, stream>>>(gmax2, hptr, scbuf);

    const int nTiles = (K0 + 15) / 16;
    int idxBlocks = (nTiles + 31) / 32;            // ~4 tiles per wave
    if (idxBlocks < 1) idxBlocks = 1;
    wect_index<<<idxBlocks, 256, 0, stream>>>(vc, wdirs, scbuf, vidx, K0);
    wect_hist<<<512, 256, 0, stream>>>(vidx, vw, ev, ew, tv, tw, hptr, gdiff,
                                       K0, K1, K2);
    wect_scan<<<1, 32, 0, stream>>>(gdiff, hptr, (float*)d_out);
}